// NNDModule_12335146074631
// MI455X (gfx1250) — compile-verified
//
#include <hip/hip_runtime.h>
#include <hip/hip_bf16.h>

typedef float v2f __attribute__((ext_vector_type(2)));
typedef float v8f __attribute__((ext_vector_type(8)));

#define BIGF 3.0e38f

// ---------------------------------------------------------------------------
// Prepass: T[b,i] = (x, y, z, x^2+y^2+z^2) as aligned float4, padded to npad
// with (0,0,0,BIGF) sentinels so the main loop needs no tail handling.
// ---------------------------------------------------------------------------
__global__ __launch_bounds__(256) void transform_points_kernel(
    const float* __restrict__ src, float4* __restrict__ dst, int n, int npad, int B)
{
    int i = blockIdx.x * blockDim.x + threadIdx.x;
    int total = B * npad;
    if (i >= total) return;
    int b = i / npad;
    int idx = i - b * npad;
    float x = 0.f, y = 0.f, z = 0.f, w = BIGF;
    if (idx < n) {
        const float* s = src + ((size_t)b * n + idx) * 3;
        x = s[0]; y = s[1]; z = s[2];
        w = x * x + y * y + z * z;
    }
    dst[i] = make_float4(x, y, z, w);
}

// ---------------------------------------------------------------------------
// Main kernel. One wave owns 32 query rows (2 strips of 16), loops over all
// targets in 16-col tiles. Per tile: ONE global_load_b64 (lane-dependent
// offset picks K0/K1 vs K2/K3 rows of B'), TWO independent WMMAs.
//   A'[m] = (-2px, -2py, -2pz, 1)      (loop-invariant, 2 VGPRs)
//   B'[n] = ( qx,   qy,   qz,  |q|^2)  (one aligned b64 load per lane)
//   C[m,:] = |p|^2                     (loop-invariant, built by 8 shuffles)
//   => D[m][n] = |p|^2 + |q|^2 - 2 p.q
// ---------------------------------------------------------------------------
__global__ __launch_bounds__(256) void chamfer_wmma_kernel(
    const float4* __restrict__ Pt, const float4* __restrict__ Qt,
    float* __restrict__ out, int NP, int NPpad, int NQpad, int B)
{
    const int lane  = threadIdx.x & 31;
    const int gwave = blockIdx.x * (blockDim.x >> 5) + (threadIdx.x >> 5);

    const int groups = NPpad >> 5;              // 32 rows per wave
    const int b = gwave / groups;
    if (b >= B) return;                         // wave-uniform: EXEC stays all-1s
    const int row0 = (gwave - b * groups) << 5;

    const int  j  = lane & 15;
    const bool hi = lane >= 16;

    const float4* Pb = Pt + (size_t)b * NPpad;
    // Per-lane base for B' tiles: column j, upper half reads floats 2..3.
    const float* Qb = (const float*)(Qt + (size_t)b * NQpad) + 4 * j + (hi ? 2 : 0);

    // ---- loop-invariant A operands and C accumulator rows ----
    float4 p0 = Pb[row0 + j];
    float4 p1 = Pb[row0 + 16 + j];

    v2f a0, a1;
    a0.x = hi ? (-2.f * p0.z) : (-2.f * p0.x);  // K0 | K2
    a0.y = hi ? 1.f           : (-2.f * p0.y);  // K1 | K3
    a1.x = hi ? (-2.f * p1.z) : (-2.f * p1.x);
    a1.y = hi ? 1.f           : (-2.f * p1.y);

    const int srcBase = hi ? 8 : 0;             // C[r] holds |p|^2 of row r+8*half
    v8f c0, c1, minv0, minv1;
#pragma unroll
    for (int r = 0; r < 8; ++r) {
        c0[r] = __shfl(p0.w, srcBase + r, 16);  // width-16: stays in each half
        c1[r] = __shfl(p1.w, srcBase + r, 16);
        minv0[r] = BIGF;
        minv1[r] = BIGF;
    }

#pragma unroll 4
    for (int col0 = 0; col0 < NQpad; col0 += 16) {
        v2f bb = *(const v2f*)(Qb + 4 * col0);  // one aligned b64 per tile

        v8f d0 = __builtin_amdgcn_wmma_f32_16x16x4_f32(
            false, a0, false, bb, (short)0, c0, false, false);
        v8f d1 = __builtin_amdgcn_wmma_f32_16x16x4_f32(
            false, a1, false, bb, (short)0, c1, false, false);

#pragma unroll
        for (int r = 0; r < 8; ++r) {
            minv0[r] = fminf(minv0[r], d0[r]);
            minv1[r] = fminf(minv1[r], d1[r]);
        }
    }

    // Min over the 16 columns: xor-shuffles never cross the 16-lane halves.
#pragma unroll
    for (int mask = 1; mask <= 8; mask <<= 1) {
#pragma unroll
        for (int r = 0; r < 8; ++r) {
            minv0[r] = fminf(minv0[r], __shfl_xor(minv0[r], mask, 32));
            minv1[r] = fminf(minv1[r], __shfl_xor(minv1[r], mask, 32));
        }
    }

    if (j == 0) {
        int mbase = row0 + (hi ? 8 : 0);
        float* ob = out + (size_t)b * NP;
#pragma unroll
        for (int r = 0; r < 8; ++r) {
            int m0 = mbase + r;
            int m1 = mbase + 16 + r;
            if (m0 < NP) ob[m0] = minv0[r];
            if (m1 < NP) ob[m1] = minv1[r];
        }
    }
}

// ---------------------------------------------------------------------------
// Fallback (no workspace): self-contained variant, one strip per wave.
// ---------------------------------------------------------------------------
__global__ __launch_bounds__(256) void chamfer_nn_min_fallback(
    const float* __restrict__ P, const float* __restrict__ Q,
    float* __restrict__ out, int NP, int NQ, int B)
{
    const int lane  = threadIdx.x & 31;
    const int gwave = blockIdx.x * (blockDim.x >> 5) + (threadIdx.x >> 5);
    const int stripsPerBatch = (NP + 15) >> 4;
    const int b = gwave / stripsPerBatch;
    if (b >= B) return;
    const int row0 = (gwave - b * stripsPerBatch) << 4;
    const int  j  = lane & 15;
    const bool hi = lane >= 16;

    const float* Pb = P + (size_t)b * NP * 3;
    const float* Qb = Q + (size_t)b * NQ * 3;

    int pi = row0 + j; if (pi >= NP) pi = NP - 1;
    float px = Pb[3 * pi], py = Pb[3 * pi + 1], pz = Pb[3 * pi + 2];
    float p2 = px * px + py * py + pz * pz;

    v2f a;
    a.x = hi ? (-2.f * pz) : (-2.f * px);
    a.y = hi ? 1.f         : (-2.f * py);

    v8f c, minv;
#pragma unroll
    for (int r = 0; r < 8; ++r) {
        c[r] = __shfl(p2, (hi ? 8 : 0) + r, 16);
        minv[r] = BIGF;
    }

#pragma unroll 2
    for (int col0 = 0; col0 < NQ; col0 += 16) {
        int qi = col0 + j;
        bool qok = qi < NQ;
        if (!qok) qi = NQ - 1;
        float qx = Qb[3 * qi], qy = Qb[3 * qi + 1], qz = Qb[3 * qi + 2];
        float q2 = qok ? (qx * qx + qy * qy + qz * qz) : BIGF;
        v2f bb;
        bb.x = hi ? qz : qx;
        bb.y = hi ? q2 : qy;
        v8f d = __builtin_amdgcn_wmma_f32_16x16x4_f32(
            false, a, false, bb, (short)0, c, false, false);
#pragma unroll
        for (int r = 0; r < 8; ++r) minv[r] = fminf(minv[r], d[r]);
    }

#pragma unroll
    for (int mask = 1; mask <= 8; mask <<= 1) {
#pragma unroll
        for (int r = 0; r < 8; ++r)
            minv[r] = fminf(minv[r], __shfl_xor(minv[r], mask, 32));
    }

    if (j == 0) {
        int mbase = row0 + (hi ? 8 : 0);
#pragma unroll
        for (int r = 0; r < 8; ++r) {
            int m = mbase + r;
            if (m < NP) out[(size_t)b * NP + m] = minv[r];
        }
    }
}

extern "C" void kernel_launch(void* const* d_in, const int* in_sizes, int n_in,
                              void* d_out, int out_size, void* d_ws, size_t ws_size,
                              hipStream_t stream) {
    const float* in1 = (const float*)d_in[0];   // [B, N, 3] f32
    const float* in2 = (const float*)d_in[1];   // [B, M, 3] f32
    float* out = (float*)d_out;                 // dist1 [B*N] ++ dist2 [B*M]

    const int B = 4;
    const int N = in_sizes[0] / (B * 3);
    const int M = in_sizes[1] / (B * 3);
    const int Npad = (N + 31) & ~31;
    const int Mpad = (M + 31) & ~31;

    size_t need = (size_t)B * (Npad + Mpad) * sizeof(float4);
    if (ws_size >= need) {
        float4* T1 = (float4*)d_ws;             // [B, Npad] transformed input1
        float4* T2 = T1 + (size_t)B * Npad;     // [B, Mpad] transformed input2

        transform_points_kernel<<<(B * Npad + 255) / 256, 256, 0, stream>>>(in1, T1, N, Npad, B);
        transform_points_kernel<<<(B * Mpad + 255) / 256, 256, 0, stream>>>(in2, T2, M, Mpad, B);

        {   // dist1
            int waves = B * (Npad >> 5);
            int blocks = (waves + 7) / 8;
            chamfer_wmma_kernel<<<blocks, 256, 0, stream>>>(T1, T2, out, N, Npad, Mpad, B);
        }
        {   // dist2
            int waves = B * (Mpad >> 5);
            int blocks = (waves + 7) / 8;
            chamfer_wmma_kernel<<<blocks, 256, 0, stream>>>(T2, T1, out + (size_t)B * N, M, Mpad, Npad, B);
        }
    } else {
        {
            int waves = B * ((N + 15) / 16);
            int blocks = (waves + 7) / 8;
            chamfer_nn_min_fallback<<<blocks, 256, 0, stream>>>(in1, in2, out, N, M, B);
        }
        {
            int waves = B * ((M + 15) / 16);
            int blocks = (waves + 7) / 8;
            chamfer_nn_min_fallback<<<blocks, 256, 0, stream>>>(in2, in1, out + (size_t)B * N, M, N, B);
        }
    }
}